// ProposalLayer_43396349558838
// MI455X (gfx1250) — compile-verified
//
#include <hip/hip_runtime.h>
#include <stdint.h>
#include <math.h>

#define BATCH_   8
#define NANCH    261888
#define PRE      6000
#define OUTC     2000
#define NBINS    16384
#define BINSHIFT 18
#define TIECAP   8192
#define SORTN    8192
#define KEEPW    188          // ceil(6000/32)
#define CHUNK    2000         // 6000 = 3 * 2000 -> uniform chunks, 2 issues/thread
#define NCHUNK   3
#define LSLOT    6            // ceil(6000/1024) boxes per lane
#define NMS_THR  0.7f

// ---------------- CDNA5 async global->LDS helpers ----------------
__device__ __forceinline__ void cdna5_async_load_b128(unsigned lds_addr, unsigned long long gaddr) {
  // VDST = LDS byte address (low 32 bits of generic shared pointer), VADDR = 64-bit global addr
  asm volatile("global_load_async_to_lds_b128 %0, %1, off" :: "v"(lds_addr), "v"(gaddr) : "memory");
}
__device__ __forceinline__ void cdna5_wait_async_le0() {
#if __has_builtin(__builtin_amdgcn_s_wait_asynccnt)
  __builtin_amdgcn_s_wait_asynccnt(0);
#else
  asm volatile("s_wait_asynccnt 0x0" ::: "memory");
#endif
}
__device__ __forceinline__ void cdna5_wait_async_le2() {
#if __has_builtin(__builtin_amdgcn_s_wait_asynccnt)
  __builtin_amdgcn_s_wait_asynccnt(2);
#else
  asm volatile("s_wait_asynccnt 0x2" ::: "memory");
#endif
}

// ---------------- kernels ----------------
__global__ void zero_kernel(int* p, int n) {
  for (int i = blockIdx.x * blockDim.x + threadIdx.x; i < n; i += gridDim.x * blockDim.x) p[i] = 0;
}

__global__ void hist_kernel(const float* __restrict__ probs, int* __restrict__ hist) {
  const int total = BATCH_ * NANCH;
  for (int g = blockIdx.x * blockDim.x + threadIdx.x; g < total; g += gridDim.x * blockDim.x) {
    float s = probs[(size_t)g * 2 + 1];            // foreground score, always >= 0
    int b = g / NANCH;
    unsigned bin = __float_as_uint(s) >> BINSHIFT; // monotonic for non-negative floats
    atomicAdd(&hist[b * NBINS + (int)bin], 1);
  }
}

__global__ void scan_kernel(const int* __restrict__ hist, int* __restrict__ tbin, int* __restrict__ above) {
  int b = blockIdx.x;
  if (threadIdx.x != 0) return;
  int cum = 0, t = 0;
  for (int bin = NBINS - 1; bin >= 0; --bin) {
    int c = hist[b * NBINS + bin];
    if (cum + c >= PRE) { t = bin; break; }
    cum += c;
  }
  tbin[b] = t;
  above[b] = cum;   // strictly-above-threshold-bin count, < PRE
}

__global__ void compact_kernel(const float* __restrict__ probs, const int* __restrict__ tbin,
                               int* selCnt, int* tieCnt,
                               unsigned long long* __restrict__ sel,
                               unsigned long long* __restrict__ tie) {
  const int total = BATCH_ * NANCH;
  for (int g = blockIdx.x * blockDim.x + threadIdx.x; g < total; g += gridDim.x * blockDim.x) {
    float s = probs[(size_t)g * 2 + 1];
    int b = g / NANCH;
    int i = g - b * NANCH;
    unsigned bits = __float_as_uint(s);
    int bin = (int)(bits >> BINSHIFT);
    int t = tbin[b];
    if (bin >= t) {
      // key: high 32 = score bits (desc), low 32 = ~idx (lower idx wins ties, matches jax top_k)
      unsigned long long key = ((unsigned long long)bits << 32) | (unsigned)(~i);
      if (bin > t) {
        int p = atomicAdd(&selCnt[b], 1);
        if (p < PRE) sel[(size_t)b * PRE + p] = key;
      } else {
        int p = atomicAdd(&tieCnt[b], 1);
        if (p < TIECAP) tie[(size_t)b * TIECAP + p] = key;
      }
    }
  }
}

__device__ __forceinline__ void bitonic_desc(unsigned long long* s, int tid, int nt) {
  for (int k = 2; k <= SORTN; k <<= 1) {
    for (int j = k >> 1; j > 0; j >>= 1) {
      for (int i = tid; i < SORTN; i += nt) {
        int ix = i ^ j;
        if (ix > i) {
          unsigned long long a = s[i], c = s[ix];
          bool up = (i & k) == 0;                 // descending overall
          if (up ? (a < c) : (a > c)) { s[i] = c; s[ix] = a; }
        }
      }
      __syncthreads();
    }
  }
}

__global__ __launch_bounds__(1024) void topk_boxes_kernel(
    const float* __restrict__ bbox, const float* __restrict__ anch,
    const int* __restrict__ tieCnt, const int* __restrict__ above,
    unsigned long long* __restrict__ sel, const unsigned long long* __restrict__ tie,
    float* __restrict__ boxesOut) {
  __shared__ unsigned long long sk[SORTN];        // 64 KB
  int b = blockIdx.x, tid = threadIdx.x, nt = blockDim.x;
  int abv = above[b];  if (abv > PRE) abv = PRE;
  int ntie = tieCnt[b]; if (ntie > TIECAP) ntie = TIECAP;

  // phase 1: sort the tie bin, take the top (PRE - above) of it
  for (int i = tid; i < SORTN; i += nt) sk[i] = (i < ntie) ? tie[(size_t)b * TIECAP + i] : 0ull;
  __syncthreads();
  bitonic_desc(sk, tid, nt);
  int need = PRE - abv; if (need < 0) need = 0; if (need > ntie) need = ntie;
  for (int r = tid; r < need; r += nt) sel[(size_t)b * PRE + abv + r] = sk[r];
  __syncthreads();

  // phase 2: sort the full 6000-selection descending (NMS order)
  int total = abv + need; if (total > PRE) total = PRE;
  for (int i = tid; i < SORTN; i += nt) sk[i] = (i < total) ? sel[(size_t)b * PRE + i] : 0ull;
  __syncthreads();
  bitonic_desc(sk, tid, nt);

  // gather anchors/deltas, apply deltas, clip, write sorted boxes
  const float4* A = (const float4*)anch;
  const float4* D = (const float4*)bbox;
  float4* O = (float4*)boxesOut;
  for (int r = tid; r < PRE; r += nt) {
    unsigned long long key = sk[r];
    unsigned idx = ~(unsigned)(key & 0xFFFFFFFFull);
    float4 o;
    if (idx < (unsigned)NANCH) {
      float4 a = A[(size_t)b * NANCH + idx];
      float4 d = D[(size_t)b * NANCH + idx];
      float h = a.z - a.x, w = a.w - a.y;
      float cy = a.x + 0.5f * h + d.x * 0.1f * h;   // RPN_BBOX_STD_DEV = {.1,.1,.2,.2}
      float cx = a.y + 0.5f * w + d.y * 0.1f * w;
      h = h * expf(d.z * 0.2f);
      w = w * expf(d.w * 0.2f);
      float y1 = cy - 0.5f * h, x1 = cx - 0.5f * w;
      float y2 = y1 + h,        x2 = x1 + w;
      o.x = fminf(fmaxf(y1, 0.f), 1.f);
      o.y = fminf(fmaxf(x1, 0.f), 1.f);
      o.z = fminf(fmaxf(y2, 0.f), 1.f);
      o.w = fminf(fmaxf(x2, 0.f), 1.f);
    } else {
      o.x = o.y = o.z = o.w = 0.f;
    }
    O[(size_t)b * PRE + r] = o;
  }
}

// Uniform issue: every thread issues exactly 2 b128 transfers -> ASYNCcnt == 2 per wave
__device__ __forceinline__ void issue_chunk_async(const float4* gb, int base, float4* buf) {
  int tid = threadIdx.x;
  int e1 = tid;                                  // tid < 1024 < CHUNK, always valid
  int e2 = tid + 1024; if (e2 > CHUNK - 1) e2 = CHUNK - 1;   // clamp: duplicate write of same data
  cdna5_async_load_b128((unsigned)(uintptr_t)&buf[e1],
                        (unsigned long long)(uintptr_t)&gb[base + e1]);
  cdna5_async_load_b128((unsigned)(uintptr_t)&buf[e2],
                        (unsigned long long)(uintptr_t)&gb[base + e2]);
}

__global__ __launch_bounds__(1024) void nms_kernel(const float* __restrict__ boxes,
                                                   unsigned* __restrict__ keepOut) {
  __shared__ float4   sChunk[2][CHUNK];  // 2 x 31.25 KB, double-buffered async DMA staging
  __shared__ unsigned sKeep[KEEPW];      // keep bitmask
  int b = blockIdx.x, tid = threadIdx.x;
  const float4* gb = (const float4*)(boxes + (size_t)b * PRE * 4);

  // lane-resident boxes: j = tid + k*1024
  float4 lb[LSLOT]; float larea[LSLOT];
  for (int k = 0; k < LSLOT; ++k) {
    int j = tid + k * 1024;
    if (j < PRE) {
      lb[k] = gb[j];
      larea[k] = (lb[k].z - lb[k].x) * (lb[k].w - lb[k].y);
    } else { lb[k].x = lb[k].y = lb[k].z = lb[k].w = 0.f; larea[k] = 0.f; }
  }
  for (int w = tid; w < KEEPW; w += 1024) sKeep[w] = 0xFFFFFFFFu;

  // prefetch chunk 0
  issue_chunk_async(gb, 0, sChunk[0]);
  __syncthreads();

  for (int c = 0; c < NCHUNK; ++c) {
    if (c + 1 < NCHUNK) {
      // prefetch next chunk into the other buffer, then wait only for the current one
      issue_chunk_async(gb, (c + 1) * CHUNK, sChunk[(c + 1) & 1]);
      cdna5_wait_async_le2();
    } else {
      cdna5_wait_async_le0();
    }
    __syncthreads();

    const float4* cur = sChunk[c & 1];
    int base = c * CHUNK;
    for (int i = base; i < base + CHUNK; ++i) {
      unsigned kw = sKeep[i >> 5];           // uniform read across the block
      if (kw & (1u << (i & 31))) {           // uniform branch
        float4 bi = cur[i - base];           // LDS broadcast
        float ai = (bi.z - bi.x) * (bi.w - bi.y);
        for (int k = 0; k < LSLOT; ++k) {
          int j = tid + k * 1024;
          if (j > i && j < PRE) {
            float yy1 = fmaxf(bi.x, lb[k].x);
            float xx1 = fmaxf(bi.y, lb[k].y);
            float yy2 = fminf(bi.z, lb[k].z);
            float xx2 = fminf(bi.w, lb[k].w);
            float inter = fmaxf(yy2 - yy1, 0.f) * fmaxf(xx2 - xx1, 0.f);
            float uni = ai + larea[k] - inter;
            float iou = (uni > 0.f) ? (inter / uni) : 0.f;
            if (iou > NMS_THR) atomicAnd(&sKeep[j >> 5], ~(1u << (j & 31)));
          }
        }
        __syncthreads();   // order clears before the next iteration's keep read
      }
    }
    __syncthreads();       // all reads of this buffer done before it is refilled (next parity)
  }
  for (int w = tid; w < KEEPW; w += 1024) keepOut[(size_t)b * KEEPW + w] = sKeep[w];
}

__global__ void gather_out_kernel(const float* __restrict__ boxes,
                                  const unsigned* __restrict__ keep,
                                  float* __restrict__ out) {
  __shared__ int pref[KEEPW + 1];
  int b = blockIdx.x, tid = threadIdx.x;
  if (tid == 0) {
    int c = 0;
    for (int w = 0; w < KEEPW; ++w) {
      unsigned bits = keep[(size_t)b * KEEPW + w];
      if (w == KEEPW - 1) bits &= 0xFFFFu;   // bits 6000..6015 invalid
      pref[w] = c;
      c += __popc(bits);
    }
    pref[KEEPW] = c;
  }
  __syncthreads();
  const float4* src = (const float4*)boxes;
  float4* dst = (float4*)out;
  for (int w = tid; w < KEEPW; w += blockDim.x) {
    unsigned bits = keep[(size_t)b * KEEPW + w];
    if (w == KEEPW - 1) bits &= 0xFFFFu;
    int r = pref[w];
    while (bits) {
      int bit = __ffs(bits) - 1;
      bits &= bits - 1;
      int j = w * 32 + bit;
      if (r < OUTC) dst[(size_t)b * OUTC + r] = src[(size_t)b * PRE + j];
      ++r;
    }
  }
  int total = pref[KEEPW]; if (total > OUTC) total = OUTC;
  for (int r = total + tid; r < OUTC; r += blockDim.x) {
    float4 z; z.x = z.y = z.z = z.w = 0.f;
    dst[(size_t)b * OUTC + r] = z;
  }
}

// ---------------- host launch ----------------
extern "C" void kernel_launch(void* const* d_in, const int* in_sizes, int n_in,
                              void* d_out, int out_size, void* d_ws, size_t ws_size,
                              hipStream_t stream) {
  (void)in_sizes; (void)n_in; (void)out_size; (void)ws_size;
  const float* probs = (const float*)d_in[0];   // (8, 261888, 2)
  const float* bbox  = (const float*)d_in[1];   // (8, 261888, 4)
  const float* anch  = (const float*)d_in[2];   // (8, 261888, 4)
  float* out = (float*)d_out;                   // (8, 2000, 4)

  char* ws = (char*)d_ws;
  size_t o = 0;
  int* hist   = (int*)(ws + o); o += (size_t)BATCH_ * NBINS * 4;   // 512 KB
  int* selCnt = (int*)(ws + o); o += BATCH_ * 4;
  int* tieCnt = (int*)(ws + o); o += BATCH_ * 4;
  int* tbin   = (int*)(ws + o); o += BATCH_ * 4;
  int* above  = (int*)(ws + o); o += BATCH_ * 4;
  o = (o + 15) & ~(size_t)15;
  unsigned long long* sel = (unsigned long long*)(ws + o); o += (size_t)BATCH_ * PRE * 8;
  o = (o + 15) & ~(size_t)15;
  unsigned long long* tie = (unsigned long long*)(ws + o); o += (size_t)BATCH_ * TIECAP * 8;
  o = (o + 15) & ~(size_t)15;
  float* boxesS = (float*)(ws + o); o += (size_t)BATCH_ * PRE * 16;  // sorted clipped boxes
  unsigned* keep = (unsigned*)(ws + o); o += (size_t)BATCH_ * KEEPW * 4;

  // hist + selCnt + tieCnt are contiguous -> one zero pass
  int zn = BATCH_ * NBINS + 2 * BATCH_;
  zero_kernel<<<512, 256, 0, stream>>>(hist, zn);
  hist_kernel<<<2048, 256, 0, stream>>>(probs, hist);
  scan_kernel<<<BATCH_, 32, 0, stream>>>(hist, tbin, above);
  compact_kernel<<<2048, 256, 0, stream>>>(probs, tbin, selCnt, tieCnt, sel, tie);
  topk_boxes_kernel<<<BATCH_, 1024, 0, stream>>>(bbox, anch, tieCnt, above, sel, tie, boxesS);
  nms_kernel<<<BATCH_, 1024, 0, stream>>>(boxesS, keep);
  gather_out_kernel<<<BATCH_, 256, 0, stream>>>(boxesS, keep, out);
}